// LightGCNLayer_49486613185210
// MI455X (gfx1250) — compile-verified
//
#include <hip/hip_runtime.h>
#include <stdint.h>

// LightGCN layer for MI455X (gfx1250, wave32).
//
// Two COO SpMMs (scatter form): out[r] += v * x[c] over random edges.
// Working set is L2-resident (38.4MB gather table + 64MB output << 192MB L2);
// bound by L2 bandwidth / f32-atomic throughput, not FLOPs -> no dense tiles
// for WMMA (random indices). CDNA5 paths used:
//   * TENSOR_LOAD_TO_LDS (Tensor Data Mover): one wave issues a descriptor-
//     driven DMA of each 4KB edge-array slab into LDS, double-buffered so the
//     DMA of chunk n+1 overlaps compute of chunk n; synced via
//     s_wait_tensorcnt + workgroup barrier. OOB reads zero-fill the tail.
//   * wave32 mapping: 16 lanes x float4 = one 64-dim embedding row (B128)
//   * non-returning global_atomic_add_f32 (STOREcnt-tracked, resolved in L2)

#define THREADS 256
#define CHUNK   1024                // edges staged per block iteration

typedef unsigned int u32x4 __attribute__((ext_vector_type(4)));
typedef unsigned int u32x8 __attribute__((ext_vector_type(8)));

__device__ __forceinline__ uint32_t lds_lo32(const void* p) {
    // Flat LDS aperture: low 32 bits of the generic pointer are the LDS byte
    // address, which is what the D# lds_addr field expects.
    return (uint32_t)(uintptr_t)p;
}

// 1-D TDM copy: `remaining` elements available from gptr (OOB -> zero-fill),
// tile of CHUNK 4-byte elements into LDS at lds_addr. 2-group descriptor.
__device__ __forceinline__ void tdm_load_1d(uint32_t lds_addr, const void* gptr,
                                            uint32_t remaining) {
    const uint64_t ga = (uint64_t)(uintptr_t)gptr;
    u32x4 g0;
    g0.x = 1u;                                            // count=1, user D#
    g0.y = lds_addr;                                      // LDS byte address
    g0.z = (uint32_t)ga;                                  // global_addr[31:0]
    g0.w = ((uint32_t)(ga >> 32) & 0x1FFFFFFu) | (2u << 30); // [56:32] | type=2
    u32x8 g1;
    g1.s0 = (2u << 16);                  // workgroup_mask=0, data_size=4B
    g1.s1 = (remaining & 0xFFFFu) << 16; // tensor_dim0[15:0]
    g1.s2 = (remaining >> 16) | (1u << 16); // tensor_dim0[31:16] | tensor_dim1=1
    g1.s3 = ((uint32_t)CHUNK) << 16;     // tensor_dim1 hi=0 | tile_dim0=CHUNK
    g1.s4 = 1u;                          // tile_dim1=1, tile_dim2=0
    g1.s5 = (uint32_t)CHUNK;             // tensor_dim0_stride[31:0]
    g1.s6 = 0u;                          // stride0 hi | stride1 lo
    g1.s7 = 0u;                          // stride1 hi
    asm volatile("tensor_load_to_lds %0, %1" :: "s"(g0), "s"(g1) : "memory");
}

__global__ void __launch_bounds__(THREADS)
zero_out_kernel(float4* __restrict__ p, long n4) {
    long i = (long)blockIdx.x * blockDim.x + threadIdx.x;
    const long stride = (long)gridDim.x * blockDim.x;
    const float4 z = make_float4(0.f, 0.f, 0.f, 0.f);
    for (; i < n4; i += stride) p[i] = z;
}

// out[r*64 + d] += v * src[c][d]; src row = (c < splitN) ? srcA[c] : srcB[c-splitN]
__global__ void __launch_bounds__(THREADS)
spmm_scatter_kernel(const int*   __restrict__ rows,
                    const int*   __restrict__ cols,
                    const float* __restrict__ vals,
                    long nnz,
                    const float* __restrict__ srcA,
                    const float* __restrict__ srcB,
                    long splitN,
                    float* __restrict__ outBase)
{
    __shared__ int   sRow[2][CHUNK];
    __shared__ int   sCol[2][CHUNK];
    __shared__ float sVal[2][CHUNK];

    const int  t       = threadIdx.x;
    const long nchunks = (nnz + CHUNK - 1) / CHUNK;
    const long c0      = blockIdx.x;
    if (c0 >= nchunks) return;

    const bool issuer = (t < 32);        // wave 0 drives the TDM

    auto stage = [&](int b, long base) {
        const uint32_t rem = (uint32_t)(nnz - base);   // OOB tail zero-fills
        tdm_load_1d(lds_lo32(&sRow[b][0]), rows + base, rem);
        tdm_load_1d(lds_lo32(&sCol[b][0]), cols + base, rem);
        tdm_load_1d(lds_lo32(&sVal[b][0]), vals + base, rem);
    };

    int cur = 0;
    if (issuer) stage(0, c0 * CHUNK);

    const int lane16 = t & 15;   // which float4 slice of the 64-dim row
    const int sub    = t >> 4;   // which edge within a group of 16

    for (long c = c0; c < nchunks; c += gridDim.x) {
        if (issuer) __builtin_amdgcn_s_wait_tensorcnt(0);  // DMA landed in LDS
        __syncthreads();                                   // visible to all waves

        const long nxt = c + (long)gridDim.x;
        if (nxt < nchunks && issuer) stage(cur ^ 1, nxt * CHUNK);

        const long base = c * CHUNK;
        if (base + CHUNK <= nnz) {
            // steady state: full chunk, no per-edge guard
            #pragma unroll 4
            for (int g = 0; g < CHUNK / 16; ++g) {
                const int   le = g * 16 + sub;
                const int   r  = sRow[cur][le];
                const long  cc = sCol[cur][le];
                const float v  = sVal[cur][le];
                const float* src = (cc < splitN)
                                 ? (srcA + cc * 64)
                                 : (srcB + (cc - splitN) * 64);
                const float4 xv =
                    *reinterpret_cast<const float4*>(src + lane16 * 4);
                float* dst = outBase + (long)r * 64 + lane16 * 4;
                atomicAdd(dst + 0, v * xv.x);
                atomicAdd(dst + 1, v * xv.y);
                atomicAdd(dst + 2, v * xv.z);
                atomicAdd(dst + 3, v * xv.w);
            }
        } else {
            // tail chunk
            for (int g = 0; g < CHUNK / 16; ++g) {
                const int le = g * 16 + sub;
                if (base + le < nnz) {
                    const int   r  = sRow[cur][le];
                    const long  cc = sCol[cur][le];
                    const float v  = sVal[cur][le];
                    const float* src = (cc < splitN)
                                     ? (srcA + cc * 64)
                                     : (srcB + (cc - splitN) * 64);
                    const float4 xv =
                        *reinterpret_cast<const float4*>(src + lane16 * 4);
                    float* dst = outBase + (long)r * 64 + lane16 * 4;
                    atomicAdd(dst + 0, v * xv.x);
                    atomicAdd(dst + 1, v * xv.y);
                    atomicAdd(dst + 2, v * xv.z);
                    atomicAdd(dst + 3, v * xv.w);
                }
            }
        }
        __syncthreads();        // everyone done reading buf[cur] before reuse
        cur ^= 1;
    }
}

extern "C" void kernel_launch(void* const* d_in, const int* in_sizes, int n_in,
                              void* d_out, int out_size, void* d_ws, size_t ws_size,
                              hipStream_t stream) {
    const float* user_emb = (const float*)d_in[0];
    const float* item_emb = (const float*)d_in[1];
    const int*   ui_rows  = (const int*)  d_in[2];
    const int*   ui_cols  = (const int*)  d_in[3];
    const float* ui_vals  = (const float*)d_in[4];
    const int*   uu_rows  = (const int*)  d_in[5];
    const int*   uu_cols  = (const int*)  d_in[6];
    const float* uu_vals  = (const float*)d_in[7];

    const long U      = (long)in_sizes[0] / 64;
    const long I      = (long)in_sizes[1] / 64;
    const long ui_nnz = (long)in_sizes[2];
    const long uu_nnz = (long)in_sizes[5];

    float* out = (float*)d_out;

    // 1) zero the accumulation target (out_size = (U+I+U)*64, multiple of 4)
    const long n4 = (long)out_size / 4;
    zero_out_kernel<<<4096, THREADS, 0, stream>>>((float4*)out, n4);

    // 2) y = A_ui * concat(user,item): rows index straight into [u_new ; i_new]
    {
        const long nchunks = (ui_nnz + CHUNK - 1) / CHUNK;
        const int  grid    = (int)(nchunks < 2048 ? nchunks : 2048);
        spmm_scatter_kernel<<<grid, THREADS, 0, stream>>>(
            ui_rows, ui_cols, ui_vals, ui_nnz,
            user_emb, item_emb, U, out);
    }

    // 3) u_social = A_uu * user_emb at offset (U+I)*64
    {
        const long nchunks = (uu_nnz + CHUNK - 1) / CHUNK;
        const int  grid    = (int)(nchunks < 2048 ? nchunks : 2048);
        spmm_scatter_kernel<<<grid, THREADS, 0, stream>>>(
            uu_rows, uu_cols, uu_vals, uu_nnz,
            user_emb, user_emb, U, out + (U + I) * 64);
    }
}